// CVRPUpperModel_38946763440477
// MI455X (gfx1250) — compile-verified
//
#include <hip/hip_runtime.h>
#include <hip/hip_bf16.h>
#include <math.h>

// CVRP attention decode step for MI455X (gfx1250), wave32.
//
// Key optimization: enc rows are rank-4 (x, y, demand, 1) combinations of
// [W_embed rows; b_embed], so all GEMMs collapse to K=4 — an exact fit for
// V_WMMA_F32_16X16X4_F32 (fp32 in/out, no precision loss, no K padding).
// Problem is then memory/latency bound (~4MB in, ~1MB out @ 23.3 TB/s).

#define B_   256
#define P_   1000
#define U_   768
#define E_   128
#define NP1  1001
#define NTILE (U_ / 16)   // 48 u-tiles per batch

typedef __attribute__((ext_vector_type(2))) float v2f;
typedef __attribute__((ext_vector_type(8))) float v8f;

// Workspace layout (floats):
#define WCK_OFF  0                         // 4x128  = M @ Wk
#define WCV_OFF  512                       // 4x128  = M @ Wv
#define WCQ_OFF  1024                      // 4x128  = M @ Wq[0:128]
#define SIGQ_OFF 1536                      // B x E  sigmoid(q)
#define NUM_OFF  (SIGQ_OFF + B_*E_)        // B x E
#define DEN_OFF  (NUM_OFF + B_*E_)         // B x E
// total = DEN_OFF + B*E = 99,840 floats = 399,360 bytes

__device__ __forceinline__ void gather_xyd(const float* __restrict__ depot_xy,
                                           const float* __restrict__ node_xy,
                                           const float* __restrict__ node_demand,
                                           int b, int idx,
                                           float& x, float& y, float& d) {
  if (idx == 0) {
    x = depot_xy[b * 2 + 0];
    y = depot_xy[b * 2 + 1];
    d = 0.0f;
  } else {
    int j = idx - 1;
    const float* p = node_xy + ((size_t)b * P_ + j) * 2;
    x = p[0];
    y = p[1];
    d = node_demand[(size_t)b * P_ + j];
  }
}

// ---------------------------------------------------------------------------
// Kernel 0: Wc = M @ W  where M = [W_embed(3xE); b_embed(1xE)]  (4 x 128 each)
// block 0 -> Wk, block 1 -> Wv, block 2 -> Wq (rows 0..127)
// ---------------------------------------------------------------------------
__global__ void __launch_bounds__(512) precomp_kernel(
    const float* __restrict__ W_embed, const float* __restrict__ b_embed,
    const float* __restrict__ Wk, const float* __restrict__ Wv,
    const float* __restrict__ Wq, float* __restrict__ ws) {
  int which = blockIdx.x;
  const float* W = (which == 0) ? Wk : ((which == 1) ? Wv : Wq);
  float* out = ws + which * 512;
  int r = threadIdx.x >> 7;      // 0..3
  int e = threadIdx.x & 127;     // 0..127
  float s = 0.0f;
  for (int j = 0; j < E_; ++j) {
    float m = (r < 3) ? W_embed[r * E_ + j] : b_embed[j];
    s += m * W[j * E_ + e];
  }
  out[r * E_ + e] = s;
}

// ---------------------------------------------------------------------------
// Kernel 1: sigq[b,e] = sigmoid( [enc_last, load] @ Wq )  via rank-4 Wcq
// ---------------------------------------------------------------------------
__global__ void __launch_bounds__(128) sigq_kernel(
    const float* __restrict__ depot_xy, const float* __restrict__ node_xy,
    const float* __restrict__ node_demand, const float* __restrict__ load,
    const int* __restrict__ current_node, const float* __restrict__ Wq,
    const float* __restrict__ wcq, float* __restrict__ sigq_out) {
  int b = blockIdx.x;
  int e = threadIdx.x;
  int cur = current_node[b];
  float x, y, d;
  gather_xyd(depot_xy, node_xy, node_demand, b, cur, x, y, d);
  float q = x * wcq[e] + y * wcq[E_ + e] + d * wcq[2 * E_ + e] + wcq[3 * E_ + e]
            + load[b] * Wq[E_ * E_ + e];  // Wq row 128 is the 'load' row
  sigq_out[b * E_ + e] = 1.0f / (1.0f + __expf(-q));
}

// ---------------------------------------------------------------------------
// Kernel 2: WMMA kernel. Per batch b:
//   k[u,:]  = (x,y,d,1)_u @ Wck   (16x16x4 fp32 WMMA)
//   v[u,:]  = (x,y,d,1)_u @ Wcv
//   num[e] += w[u] * exp(k[u,e]) * v[u,e];  den[e] += w[u] * exp(k[u,e])
// 8 waves per block; wave w owns e-columns [16w, 16w+16). Loop-invariant B
// fragments; A fragments read from an LDS-staged (x,y,d,1) table.
// ---------------------------------------------------------------------------
__global__ void __launch_bounds__(256) kv_attn_kernel(
    const float* __restrict__ depot_xy, const float* __restrict__ node_xy,
    const float* __restrict__ node_demand, const float* __restrict__ cur_dist,
    const float* __restrict__ ninf_mask, const float* __restrict__ log_scale,
    const float* __restrict__ alpha_attn, const int* __restrict__ unvisited,
    const float* __restrict__ wck, const float* __restrict__ wcv,
    float* __restrict__ num_out, float* __restrict__ den_out) {
  __shared__ float4 xyzd[U_];   // (x, y, d, 1) per unvisited row
  __shared__ float  wexp[U_];   // exp(-log_scale*alpha_attn*dist + ninf)

  int b = blockIdx.x;
  int tid = threadIdx.x;
  float scale = -log_scale[0] * alpha_attn[0];

  for (int u = tid; u < U_; u += 256) {
    int idx = unvisited[b * U_ + u];
    float x, y, d;
    gather_xyd(depot_xy, node_xy, node_demand, b, idx, x, y, d);
    xyzd[u] = make_float4(x, y, d, 1.0f);
    wexp[u] = __expf(scale * cur_dist[b * U_ + u] + ninf_mask[b * U_ + u]);
  }
  __syncthreads();

  int wave = tid >> 5;
  int lane = tid & 31;
  int half = lane >> 4;          // 0: K=0,1   1: K=2,3
  int nl   = lane & 15;          // column within 16-wide tile
  int n    = (wave << 4) + nl;   // global e column

  // B fragment (4x16 fp32): VGPR g holds row K = g + 2*half, column n.
  v2f bk, bv;
  bk.x = wck[(2 * half + 0) * E_ + n];
  bk.y = wck[(2 * half + 1) * E_ + n];
  bv.x = wcv[(2 * half + 0) * E_ + n];
  bv.y = wcv[(2 * half + 1) * E_ + n];

  float nacc[8], dacc[8];
#pragma unroll
  for (int r = 0; r < 8; ++r) { nacc[r] = 0.0f; dacc[r] = 0.0f; }

  for (int t = 0; t < NTILE; ++t) {
    // A fragment (16x4 fp32): lane (m=nl, half) holds K = 2*half + {0,1}
    const float* a4 = (const float*)&xyzd[t * 16 + nl];
    v2f a;
    a.x = a4[2 * half + 0];
    a.y = a4[2 * half + 1];

    v8f ck = {};
    v8f cv = {};
    ck = __builtin_amdgcn_wmma_f32_16x16x4_f32(false, a, false, bk,
                                               (short)0, ck, false, false);
    cv = __builtin_amdgcn_wmma_f32_16x16x4_f32(false, a, false, bv,
                                               (short)0, cv, false, false);

    // C/D layout: VGPR r holds M = r + 8*half, N = lane&15.
#pragma unroll
    for (int r = 0; r < 8; ++r) {
      int u = t * 16 + r + 8 * half;
      float wv = wexp[u];
      float ek = __expf(ck[r]);
      nacc[r] += wv * ek * cv[r];
      dacc[r] += wv * ek;
    }
  }

  float sn = 0.0f, sd = 0.0f;
#pragma unroll
  for (int r = 0; r < 8; ++r) { sn += nacc[r]; sd += dacc[r]; }
  // combine the two half-lane partial sums (same column n, disjoint u rows)
  sn += __shfl_xor(sn, 16, 32);
  sd += __shfl_xor(sd, 16, 32);
  if (lane < 16) {
    num_out[b * E_ + n] = sn;
    den_out[b * E_ + n] = sd;
  }
}

// ---------------------------------------------------------------------------
// Kernel 3: aafm = sigmoid(q)*num/den; score via rank-4 dot; tanh-clip
// softmax over U; argmax (first-max); scatter probs into (P+1)-wide row.
// ---------------------------------------------------------------------------
__global__ void __launch_bounds__(256) score_softmax_kernel(
    const float* __restrict__ depot_xy, const float* __restrict__ node_xy,
    const float* __restrict__ node_demand, const float* __restrict__ cur_dist,
    const float* __restrict__ ninf_mask, const float* __restrict__ log_scale,
    const float* __restrict__ alpha_com, const float* __restrict__ W_embed,
    const float* __restrict__ b_embed, const int* __restrict__ unvisited,
    const float* __restrict__ sigq, const float* __restrict__ num,
    const float* __restrict__ den, float* __restrict__ out_up,
    float* __restrict__ out_ts, float* __restrict__ out_ss) {
  __shared__ float aafm[E_];
  __shared__ float red[256];
  __shared__ float cvec[4];
  __shared__ float bestp_s[256];
  __shared__ int   bestu_s[256];

  int b = blockIdx.x;
  int tid = threadIdx.x;

  if (tid < E_) {
    int o = b * E_ + tid;
    aafm[tid] = sigq[o] * num[o] / den[o];
  }
  __syncthreads();

  // cvec[c] = aafm . M[c]  with M = [W0; W1; W2; b_embed]
  for (int c = 0; c < 4; ++c) {
    float p = 0.0f;
    if (tid < E_) {
      float m = (c < 3) ? W_embed[c * E_ + tid] : b_embed[tid];
      p = aafm[tid] * m;
    }
    red[tid] = p;
    __syncthreads();
    for (int s = 128; s > 0; s >>= 1) {
      if (tid < s) red[tid] += red[tid + s];
      __syncthreads();
    }
    if (tid == 0) cvec[c] = red[0];
    __syncthreads();
  }
  float c0 = cvec[0], c1 = cvec[1], c2 = cvec[2], c3 = cvec[3];

  float scale = -log_scale[0] * alpha_com[0];
  const float inv_sqrtE = 0.0883883476483184f;  // 1/sqrt(128)

  float z[3];
  int   gidx[3];
  float lmax = -INFINITY;
#pragma unroll
  for (int i = 0; i < 3; ++i) {
    int u = tid + i * 256;
    int idx = unvisited[b * U_ + u];
    gidx[i] = idx;
    float x, y, d;
    gather_xyd(depot_xy, node_xy, node_demand, b, idx, x, y, d);
    float s = (x * c0 + y * c1 + d * c2 + c3) * inv_sqrtE
              + scale * cur_dist[b * U_ + u];
    float zz = 10.0f * tanhf(s) + ninf_mask[b * U_ + u];
    z[i] = zz;
    lmax = fmaxf(lmax, zz);
  }

  // block max
  red[tid] = lmax;
  __syncthreads();
  for (int s = 128; s > 0; s >>= 1) {
    if (tid < s) red[tid] = fmaxf(red[tid], red[tid + s]);
    __syncthreads();
  }
  float M = red[0];
  __syncthreads();

  float p[3];
  float lsum = 0.0f;
#pragma unroll
  for (int i = 0; i < 3; ++i) {
    p[i] = __expf(z[i] - M);
    lsum += p[i];
  }
  red[tid] = lsum;
  __syncthreads();
  for (int s = 128; s > 0; s >>= 1) {
    if (tid < s) red[tid] += red[tid + s];
    __syncthreads();
  }
  float invS = 1.0f / red[0];
  __syncthreads();

  // zero the full (P+1)-wide output row, then scatter probs
  for (int j = tid; j < NP1; j += 256) out_up[(size_t)b * NP1 + j] = 0.0f;
  __syncthreads();

  float bp = -1.0f;
  int   bu = 0;
#pragma unroll
  for (int i = 0; i < 3; ++i) {
    float pn = p[i] * invS;
    out_up[(size_t)b * NP1 + gidx[i]] = pn;
    int u = tid + i * 256;
    if (pn > bp) { bp = pn; bu = u; }   // ascending u within thread -> first max
  }
  bestp_s[tid] = bp;
  bestu_s[tid] = bu;
  __syncthreads();
  for (int s = 128; s > 0; s >>= 1) {
    if (tid < s) {
      float p2 = bestp_s[tid + s];
      int   u2 = bestu_s[tid + s];
      if (p2 > bestp_s[tid] || (p2 == bestp_s[tid] && u2 < bestu_s[tid])) {
        bestp_s[tid] = p2;
        bestu_s[tid] = u2;
      }
    }
    __syncthreads();
  }
  if (tid == 0) {
    int sel = bestu_s[0];
    out_ts[b] = (float)unvisited[b * U_ + sel];  // true_selected (as float)
    out_ss[b] = bestp_s[0];                      // selected_score
  }
}

// ---------------------------------------------------------------------------
extern "C" void kernel_launch(void* const* d_in, const int* in_sizes, int n_in,
                              void* d_out, int out_size, void* d_ws,
                              size_t ws_size, hipStream_t stream) {
  const float* depot_xy    = (const float*)d_in[0];
  const float* node_xy     = (const float*)d_in[1];
  const float* node_demand = (const float*)d_in[2];
  const float* load        = (const float*)d_in[3];
  const float* cur_dist    = (const float*)d_in[4];
  const float* ninf_mask   = (const float*)d_in[5];
  const float* log_scale   = (const float*)d_in[6];
  const float* W_embed     = (const float*)d_in[7];
  const float* b_embed     = (const float*)d_in[8];
  const float* Wq          = (const float*)d_in[9];
  const float* Wk          = (const float*)d_in[10];
  const float* Wv          = (const float*)d_in[11];
  const float* alpha_attn  = (const float*)d_in[12];
  const float* alpha_com   = (const float*)d_in[13];
  const int*   current_node = (const int*)d_in[14];
  const int*   unvisited    = (const int*)d_in[15];

  float* ws = (float*)d_ws;
  float* out = (float*)d_out;
  float* out_up = out;                    // B x 1001
  float* out_ts = out + (size_t)B_ * NP1; // B
  float* out_ss = out_ts + B_;            // B

  precomp_kernel<<<3, 512, 0, stream>>>(W_embed, b_embed, Wk, Wv, Wq, ws);

  sigq_kernel<<<B_, 128, 0, stream>>>(depot_xy, node_xy, node_demand, load,
                                      current_node, Wq, ws + WCQ_OFF,
                                      ws + SIGQ_OFF);

  kv_attn_kernel<<<B_, 256, 0, stream>>>(
      depot_xy, node_xy, node_demand, cur_dist, ninf_mask, log_scale,
      alpha_attn, unvisited, ws + WCK_OFF, ws + WCV_OFF, ws + NUM_OFF,
      ws + DEN_OFF);

  score_softmax_kernel<<<B_, 256, 0, stream>>>(
      depot_xy, node_xy, node_demand, cur_dist, ninf_mask, log_scale,
      alpha_com, W_embed, b_embed, unvisited, ws + SIGQ_OFF, ws + NUM_OFF,
      ws + DEN_OFF, out_up, out_ts, out_ss);
}